// MotionCorrectionLoss_26998164423038
// MI455X (gfx1250) — compile-verified
//
#include <hip/hip_runtime.h>
#include <cstdint>
#include <cstddef>

#ifndef __has_builtin
#define __has_builtin(x) 0
#endif

#if __has_builtin(__builtin_amdgcn_global_load_async_to_lds_b128)
#define USE_ASYNC_LDS 1
#else
#define USE_ASYNC_LDS 0
#endif

typedef int v4i __attribute__((ext_vector_type(4)));

namespace {
constexpr int Bc   = 128;          // batch
constexpr int Sc   = 1024;         // sequence
constexpr int CH   = 132;          // channels per (b,s) = 22 joints * 6
constexpr int TILE = 128;          // s-steps per block
constexpr int NT   = Sc / TILE;    // tiles per batch row
constexpr int NB   = Bc * NT;      // number of blocks for kernel 1
constexpr int BLK  = 128;          // threads per block (== TILE)
constexpr int STR  = 140;          // padded LDS row stride in floats (16B aligned, low-conflict)
constexpr int PSTR = 12;           // per-block partial stride in floats (9 used)
}

// ---------------- async global -> LDS helper ----------------
__device__ __forceinline__ void cp16_g2l(const float* __restrict__ gsrc, float* ldst) {
#if USE_ASYNC_LDS
  __builtin_amdgcn_global_load_async_to_lds_b128((v4i*)gsrc, (v4i*)ldst, 0, 0);
#else
  *(float4*)ldst = *(const float4*)gsrc;
#endif
}

__device__ __forceinline__ void wait_async_lds() {
#if USE_ASYNC_LDS
#if __has_builtin(__builtin_amdgcn_s_wait_asynccnt)
  __builtin_amdgcn_s_wait_asynccnt(0);
#else
  asm volatile("s_wait_asynccnt 0" ::: "memory");
#endif
#endif
}

// ---------------- 6d -> 3x3 rotation (row-major, columns = b1,b2,b3) ----------------
__device__ __forceinline__ void rot6d_to_mat(const float* __restrict__ d, float* __restrict__ R) {
  const float ax = d[0], ay = d[1], az = d[2];
  const float bx = d[3], by = d[4], bz = d[5];
  const float inv1 = 1.0f / fmaxf(sqrtf(ax*ax + ay*ay + az*az), 1e-12f);
  const float b1x = ax*inv1, b1y = ay*inv1, b1z = az*inv1;
  const float dp = b1x*bx + b1y*by + b1z*bz;
  const float ux = bx - dp*b1x, uy = by - dp*b1y, uz = bz - dp*b1z;
  const float inv2 = 1.0f / fmaxf(sqrtf(ux*ux + uy*uy + uz*uz), 1e-12f);
  const float b2x = ux*inv2, b2y = uy*inv2, b2z = uz*inv2;
  const float b3x = b1y*b2z - b1z*b2y;
  const float b3y = b1z*b2x - b1x*b2z;
  const float b3z = b1x*b2y - b1y*b2x;
  R[0]=b1x; R[1]=b2x; R[2]=b3x;
  R[3]=b1y; R[4]=b2y; R[5]=b3y;
  R[6]=b1z; R[7]=b2z; R[8]=b3z;
}

__device__ __forceinline__ void mat33_mul(const float* __restrict__ A,
                                          const float* __restrict__ B,
                                          float* __restrict__ C) {
#pragma unroll
  for (int r = 0; r < 3; ++r) {
#pragma unroll
    for (int c = 0; c < 3; ++c)
      C[3*r+c] = A[3*r+0]*B[c] + A[3*r+1]*B[3+c] + A[3*r+2]*B[6+c];
  }
}

// ---------------- kernel 1: per-tile losses ----------------
__global__ __launch_bounds__(BLK)
void motion_loss_tile_kernel(const float* __restrict__ root,   // (B,S,2,6)
                             const float* __restrict__ body,   // (B,S,20,6)
                             const float* __restrict__ tgt,    // (B,S,132)
                             const float* __restrict__ mask,   // (B,S)
                             const float* __restrict__ offs,   // (22,3)
                             float* __restrict__ part)         // (NB, PSTR)
{
  __shared__ float ldsP[(TILE + 2) * STR];   // pred tile (+2 halo rows)
  __shared__ float ldsT[(TILE + 2) * STR];   // target tile

  const int tile = blockIdx.x % NT;
  const int b    = blockIdx.x / NT;
  const int s0   = tile * TILE;

  int s_begin, row_off, nrows;
  if (tile == 0) { s_begin = 0;      row_off = 2; nrows = TILE;     }
  else           { s_begin = s0 - 2; row_off = 0; nrows = TILE + 2; }

  const float* rootB = root + ((size_t)b * Sc + s_begin) * 12;
  const float* bodyB = body + ((size_t)b * Sc + s_begin) * 120;
  const float* tgtB  = tgt  + ((size_t)b * Sc + s_begin) * CH;

  // ---- stage tiles into LDS (contiguous global reads, async B128 granules).
  // Incremental row/chunk counters avoid per-iteration div/mod (stride = BLK).
  {
    int row = threadIdx.x / 3, k4 = threadIdx.x % 3;            // pred channels 0..11
    for (int g = threadIdx.x; g < nrows * 3; g += BLK) {
      cp16_g2l(rootB + 4 * g, &ldsP[(row_off + row) * STR + 4 * k4]);
      row += 42; k4 += 2;                                       // 128 = 42*3 + 2
      if (k4 >= 3) { k4 -= 3; row += 1; }
    }
  }
  {
    int row = threadIdx.x / 30, k4 = threadIdx.x % 30;          // pred channels 12..131
    for (int g = threadIdx.x; g < nrows * 30; g += BLK) {
      cp16_g2l(bodyB + 4 * g, &ldsP[(row_off + row) * STR + 12 + 4 * k4]);
      row += 4; k4 += 8;                                        // 128 = 4*30 + 8
      if (k4 >= 30) { k4 -= 30; row += 1; }
    }
  }
  {
    int row = threadIdx.x / 33, k4 = threadIdx.x % 33;          // target channels 0..131
    for (int g = threadIdx.x; g < nrows * 33; g += BLK) {
      cp16_g2l(tgtB + 4 * g, &ldsT[(row_off + row) * STR + 4 * k4]);
      row += 3; k4 += 29;                                       // 128 = 3*33 + 29
      if (k4 >= 33) { k4 -= 33; row += 1; }
    }
  }
  wait_async_lds();
  __syncthreads();

  // ---- per-thread: one s-step ----
  const int  r  = s0 + threadIdx.x;          // global s index
  const int  lr = threadIdx.x + 2;           // LDS row
  const float w = mask[(size_t)b * Sc + r];
  const bool hasV = (r >= 1), hasA = (r >= 2);

  float se_rv = 0.f, se_rr = 0.f, se_pr = 0.f, sv = 0.f, sa = 0.f;
  {
    const float* r0P = &ldsP[lr * STR];
    const float* r0T = &ldsT[lr * STR];
    const float* r1P = &ldsP[(lr - 1) * STR];
    const float* r1T = &ldsT[(lr - 1) * STR];
    const float* r2P = &ldsP[(lr - 2) * STR];
    const float* r2T = &ldsT[(lr - 2) * STR];
#pragma unroll
    for (int k4 = 0; k4 < 33; ++k4) {
      const float4 p = *(const float4*)(r0P + 4 * k4);
      const float4 t = *(const float4*)(r0T + 4 * k4);
      const float ex = p.x - t.x, ey = p.y - t.y, ez = p.z - t.z, ew = p.w - t.w;
      const float s4 = ex*ex + ey*ey + ez*ez + ew*ew;
      if      (k4 == 0) se_rv += ex*ex + ey*ey + ez*ez;  // joint0 c0..2 (root_vel)
      else if (k4 == 1) se_rr += ez*ez + ew*ew;          // ch 6,7 (root_rot)
      else if (k4 == 2) se_rr += s4;                     // ch 8..11
      else              se_pr += s4;                     // ch 12..131 (pose_rot)
      if (hasV) {
        const float4 p1 = *(const float4*)(r1P + 4 * k4);
        const float4 t1 = *(const float4*)(r1T + 4 * k4);
        const float e1x = p1.x - t1.x, e1y = p1.y - t1.y, e1z = p1.z - t1.z, e1w = p1.w - t1.w;
        const float vx = ex - e1x, vy = ey - e1y, vz = ez - e1z, vw = ew - e1w;
        sv += vx*vx + vy*vy + vz*vz + vw*vw;
        if (hasA) {
          const float4 p2 = *(const float4*)(r2P + 4 * k4);
          const float4 t2 = *(const float4*)(r2T + 4 * k4);
          const float e2x = p2.x - t2.x, e2y = p2.y - t2.y, e2z = p2.z - t2.z, e2w = p2.w - t2.w;
          const float axx = vx - (e1x - e2x), ayy = vy - (e1y - e2y);
          const float azz = vz - (e1z - e2z), aww = vw - (e1w - e2w);
          sa += axx*axx + ayy*ayy + azz*azz + aww*aww;
        }
      }
    }
  }

  // ---- forward kinematics (6 chains, saved state at joint 9) ----
  float spos = 0.f;
  {
    const float* rowP = &ldsP[lr * STR];
    const float* rowT = &ldsT[lr * STR];
    float Rp[9] = {1,0,0, 0,1,0, 0,0,1};
    float Rt[9] = {1,0,0, 0,1,0, 0,0,1};
    float Rn[9], M[9];
    float ppx = 0, ppy = 0, ppz = 0, ptx = 0, pty = 0, ptz = 0;
    float R9p[9] = {1,0,0,0,1,0,0,0,1}, R9t[9] = {1,0,0,0,1,0,0,0,1};
    float p9px = 0, p9py = 0, p9pz = 0, p9tx = 0, p9ty = 0, p9tz = 0;
    const int WJ[21] = {1,4,7,10, 2,5,8,11, 3,6,9, 12,15, 13,16,18,20, 14,17,19,21};
    const int WR[21] = {1,0,0,0,  1,0,0,0,  1,0,0, 2,0,   2,0,0,0,     2,0,0,0};
#pragma unroll
    for (int idx = 0; idx < 21; ++idx) {
      const int j = WJ[idx];
      if (WR[idx] == 1) {
#pragma unroll
        for (int q = 0; q < 9; ++q) { Rp[q] = (q%4==0)?1.f:0.f; Rt[q] = (q%4==0)?1.f:0.f; }
        ppx = ppy = ppz = 0.f; ptx = pty = ptz = 0.f;
      } else if (WR[idx] == 2) {
#pragma unroll
        for (int q = 0; q < 9; ++q) { Rp[q] = R9p[q]; Rt[q] = R9t[q]; }
        ppx = p9px; ppy = p9py; ppz = p9pz; ptx = p9tx; pty = p9ty; ptz = p9tz;
      }
      const float ox = offs[3*j+0], oy = offs[3*j+1], oz = offs[3*j+2];
      const float npx = ppx + Rp[0]*ox + Rp[1]*oy + Rp[2]*oz;
      const float npy = ppy + Rp[3]*ox + Rp[4]*oy + Rp[5]*oz;
      const float npz = ppz + Rp[6]*ox + Rp[7]*oy + Rp[8]*oz;
      const float ntx = ptx + Rt[0]*ox + Rt[1]*oy + Rt[2]*oz;
      const float nty = pty + Rt[3]*ox + Rt[4]*oy + Rt[5]*oz;
      const float ntz = ptz + Rt[6]*ox + Rt[7]*oy + Rt[8]*oz;
      const float dx = npx - ntx, dy = npy - nty, dz = npz - ntz;
      spos += dx*dx + dy*dy + dz*dz;
      rot6d_to_mat(rowP + 6*j, M); mat33_mul(Rp, M, Rn);
#pragma unroll
      for (int q = 0; q < 9; ++q) Rp[q] = Rn[q];
      rot6d_to_mat(rowT + 6*j, M); mat33_mul(Rt, M, Rn);
#pragma unroll
      for (int q = 0; q < 9; ++q) Rt[q] = Rn[q];
      ppx = npx; ppy = npy; ppz = npz; ptx = ntx; pty = nty; ptz = ntz;
      if (j == 9) {
#pragma unroll
        for (int q = 0; q < 9; ++q) { R9p[q] = Rp[q]; R9t[q] = Rt[q]; }
        p9px = ppx; p9py = ppy; p9pz = ppz; p9tx = ptx; p9ty = pty; p9tz = ptz;
      }
    }
  }

  // ---- deterministic block reduction: shuffle within wave32, LDS across waves ----
  float acc[9];
  acc[0] = w * se_rv; acc[1] = w * se_rr; acc[2] = w * se_pr; acc[3] = w * spos;
  acc[4] = hasV ? w * sv : 0.f;
  acc[5] = hasA ? w * sa : 0.f;
  acc[6] = w;
  acc[7] = hasV ? w : 0.f;
  acc[8] = hasA ? w : 0.f;
#pragma unroll
  for (int off = 16; off > 0; off >>= 1) {
#pragma unroll
    for (int i = 0; i < 9; ++i) acc[i] += __shfl_down(acc[i], off, 32);
  }
  __syncthreads();                       // all LDS tile reads done; safe to reuse
  const int wid  = threadIdx.x >> 5;
  const int lane = threadIdx.x & 31;
  if (lane == 0) {
#pragma unroll
    for (int i = 0; i < 9; ++i) ldsP[wid * 9 + i] = acc[i];
  }
  __syncthreads();
  if (threadIdx.x < 9) {
    const int i = threadIdx.x;
    part[(size_t)blockIdx.x * PSTR + i] =
        ldsP[i] + ldsP[9 + i] + ldsP[18 + i] + ldsP[27 + i];
  }
}

// ---------------- kernel 2: fixed-order final reduction + loss assembly ----------------
__global__ __launch_bounds__(256)
void motion_loss_final_kernel(const float* __restrict__ part, float* __restrict__ out) {
  __shared__ double red[256];
  __shared__ double finals[9];
  const int t = threadIdx.x;
  double comp[9];
#pragma unroll
  for (int c = 0; c < 9; ++c) {
    double s = 0.0;
    for (int bk = t; bk < NB; bk += 256) s += (double)part[(size_t)bk * PSTR + c];
    comp[c] = s;
  }
  for (int c = 0; c < 9; ++c) {
    red[t] = comp[c];
    __syncthreads();
    for (int off = 128; off > 0; off >>= 1) {
      if (t < off) red[t] += red[t + off];
      __syncthreads();
    }
    if (t == 0) finals[c] = red[0];
    __syncthreads();
  }
  if (t == 0) {
    const double vc  = finals[6] + 1e-8;
    const double vcv = finals[7] + 1e-8;
    const double vca = finals[8] + 1e-8;
    const double rv = finals[0] / (vc  * 3.0);
    const double rr = finals[1] / (vc  * 6.0);
    const double pr = finals[2] / (vc  * 120.0);   // 6*20
    const double pp = finals[3] / (vc  * 66.0);    // 3*22
    const double sm = finals[4] / (vcv * 132.0);   // 6*22
    const double ac = finals[5] / (vca * 132.0);
    const double tot = rv + rr + pr + 2.0*pp + 10.0*sm + 20.0*ac;
    out[0] = (float)tot; out[1] = (float)rv; out[2] = (float)rr; out[3] = (float)pr;
    out[4] = (float)pp;  out[5] = (float)sm; out[6] = (float)ac;
  }
}

extern "C" void kernel_launch(void* const* d_in, const int* in_sizes, int n_in,
                              void* d_out, int out_size, void* d_ws, size_t ws_size,
                              hipStream_t stream) {
  (void)in_sizes; (void)n_in; (void)out_size; (void)ws_size;
  const float* root = (const float*)d_in[0];   // (128,1024,2,6)
  const float* body = (const float*)d_in[1];   // (128,1024,20,6)
  const float* tgt  = (const float*)d_in[2];   // (128,1024,132)
  const float* mask = (const float*)d_in[3];   // (128,1024)
  const float* offs = (const float*)d_in[4];   // (22,3)
  float* part = (float*)d_ws;                  // NB * PSTR floats = 48 KB
  float* outp = (float*)d_out;                 // 7 floats

  motion_loss_tile_kernel<<<NB, BLK, 0, stream>>>(root, body, tgt, mask, offs, part);
  motion_loss_final_kernel<<<1, 256, 0, stream>>>(part, outp);
}